// LatentScaleSelectionHead_42803644072399
// MI455X (gfx1250) — compile-verified
//
#include <hip/hip_runtime.h>

// ---------------------------------------------------------------------------
// LatentScaleSelectionHead for MI455X (gfx1250, wave32, WMMA)
//
// joint[:, -4:] needs only 16 gathered rows =>
//   G    = D11*(A11+I) rows[pos]                      (16 x 1125)
//   H    = colscale(X_{k+1}, 1/rowsum_k)
//   X_k  = H @ A_k + H                                 k = 10, 9, 8
// Heavy op = skinny f32 GEMM -> V_WMMA_F32_16X16X4_F32, one wave / 16-col tile.
// Branch-free inner loop (clamped N, split K tail), dual accumulators to
// overlap the XDL RAW latency. 40MB of attn blocks 8..11 stays L2-resident.
// ---------------------------------------------------------------------------

#define Bsz        2
#define NUM_POINTS 100
#define Nn         1125
#define NPAD       1128        // multiple of 4; cols/rows >= 1125 are zero
#define NUM_GT     16
#define CAM_THR    0.2f

typedef __attribute__((ext_vector_type(2))) float v2f;
typedef __attribute__((ext_vector_type(8))) float v8f;

// invrs[b][blk-8][row] = 1/(1 + sum_j attn[blk][b][row][j]); rows >= 1125 -> 0
__global__ void __launch_bounds__(32) k_rowsum(const float* __restrict__ attn,
                                               float* __restrict__ invrs) {
  int row = blockIdx.x;           // 0..NPAD-1
  int blkoff = blockIdx.y;        // 0..3 -> block 8+blkoff
  int b = blockIdx.z;
  int lane = threadIdx.x;
  float out = 0.0f;
  if (row < Nn) {
    const float* rp = attn + (((size_t)(8 + blkoff) * Bsz + b) * Nn + row) * (size_t)Nn;
    float s = 0.0f;
    for (int j = lane; j < Nn; j += 32) s += rp[j];
    for (int m = 16; m >= 1; m >>= 1) s += __shfl_xor(s, m, 32);
    out = 1.0f / (s + 1.0f);
  }
  if (lane == 0) invrs[((size_t)b * 4 + blkoff) * NPAD + row] = out;
}

// X[b][3][g][:] = normalized gathered rows of aug11 (cols >= 1125 zeroed)
__global__ void k_gather(const float* __restrict__ attn, const int* __restrict__ pos,
                         const float* __restrict__ invrs, float* __restrict__ X) {
  int g = blockIdx.x, b = blockIdx.y;
  int row = Nn - NUM_POINTS + pos[b * NUM_GT + g];
  float sc = invrs[((size_t)b * 4 + 3) * NPAD + row];
  const float* rp = attn + (((size_t)11 * Bsz + b) * Nn + row) * (size_t)Nn;
  float* xr = X + ((size_t)(b * 4 + 3) * NUM_GT + g) * NPAD;
  for (int n = threadIdx.x; n < NPAD; n += blockDim.x) {
    float v = 0.0f;
    if (n < Nn) v = (rp[n] + (n == row ? 1.0f : 0.0f)) * sc;
    xr[n] = v;
  }
}

// H[b][m][k] = X[b][s_src][m][k] * invrs[b][blk-8][k]  (pads stay 0: 0*0)
__global__ void k_scale(const float* __restrict__ X, const float* __restrict__ invrs,
                        float* __restrict__ H, int s_src, int blk) {
  int m = blockIdx.x, b = blockIdx.y;
  const float* xr = X + ((size_t)(b * 4 + s_src) * NUM_GT + m) * NPAD;
  const float* inv = invrs + ((size_t)b * 4 + (blk - 8)) * NPAD;
  float* hr = H + ((size_t)b * NUM_GT + m) * NPAD;
  for (int k = threadIdx.x; k < NPAD; k += blockDim.x) hr[k] = xr[k] * inv[k];
}

// One wave per 16-wide N tile: Xdst = H @ A + H. Branch-free inner loop.
__global__ void __launch_bounds__(32) k_step(const float* __restrict__ attn,
                                             const float* __restrict__ H,
                                             float* __restrict__ Xout,
                                             int blk, int s_dst) {
  int ntile = blockIdx.x;         // 0..70
  int b = blockIdx.y;
  int lane = threadIdx.x;
  int half = lane >> 4;           // WMMA lane halves
  int l16 = lane & 15;
  const float* Hb = H + (size_t)b * NUM_GT * NPAD;
  float* Xdst = Xout + (size_t)(b * 4 + s_dst) * NUM_GT * NPAD;
  const float* A = attn + ((size_t)blk * Bsz + b) * (size_t)Nn * Nn;
  int n = ntile * 16 + l16;
  int nc = (n < Nn) ? n : (Nn - 1);                 // clamp: garbage cols never stored
  const float* hrow = Hb + (size_t)l16 * NPAD;      // A-frag row m == l16 (both halves)

  v8f c0 = {}, c1 = {};
  // K = 0..1119, fully in-bounds, two independent accumulator chains
  for (int kb = 0; kb < 1120; kb += 8) {
    int ka = kb + 2 * half;                         // lanes 0-15: K,K+1 ; 16-31: K+2,K+3
    v2f a0 = *(const v2f*)(hrow + ka);
    v2f b0; b0.x = A[(size_t)ka * Nn + nc];
            b0.y = A[(size_t)(ka + 1) * Nn + nc];
    c0 = __builtin_amdgcn_wmma_f32_16x16x4_f32(false, a0, false, b0, (short)0, c0,
                                               false, false);
    int kc = ka + 4;
    v2f a1 = *(const v2f*)(hrow + kc);
    v2f b1; b1.x = A[(size_t)kc * Nn + nc];
            b1.y = A[(size_t)(kc + 1) * Nn + nc];
    c1 = __builtin_amdgcn_wmma_f32_16x16x4_f32(false, a1, false, b1, (short)0, c1,
                                               false, false);
  }
  {// K = 1120..1123 (in-bounds)
    int ka = 1120 + 2 * half;
    v2f a0 = *(const v2f*)(hrow + ka);
    v2f b0; b0.x = A[(size_t)ka * Nn + nc];
            b0.y = A[(size_t)(ka + 1) * Nn + nc];
    c0 = __builtin_amdgcn_wmma_f32_16x16x4_f32(false, a0, false, b0, (short)0, c0,
                                               false, false);
  }
  {// K = 1124 tail: only lane-half 0's first element is real; rest zero (cndmask)
    v2f a1 = *(const v2f*)(hrow + 1124 + 2 * half); // H pads (>=1125) are 0
    float vA = A[(size_t)1124 * Nn + nc];           // valid address for all lanes
    v2f b1; b1.x = half ? 0.0f : vA;  b1.y = 0.0f;
    c1 = __builtin_amdgcn_wmma_f32_16x16x4_f32(false, a1, false, b1, (short)0, c1,
                                               false, false);
  }

  // epilogue: combine chains, + identity term; zero pad columns of last tile
#pragma unroll
  for (int vr = 0; vr < 8; ++vr) {
    int m = vr + 8 * half;        // C/D layout: lanes 16-31 hold M+8
    if (n < Nn) {
      Xdst[(size_t)m * NPAD + n] = c0[vr] + c1[vr] + Hb[(size_t)m * NPAD + n];
    } else if (n < NPAD) {
      Xdst[(size_t)m * NPAD + n] = 0.0f;
    }
  }
}

// One block per (bg, s) 32x32 map: normalize, threshold, 11x11 close, bbox
__global__ void __launch_bounds__(1024) k_post(const float* __restrict__ X,
                                               float* __restrict__ out) {
  __shared__ float sa[1024];
  __shared__ float sb[1024];
  int q = blockIdx.x;             // bg*4 + s
  int b = q >> 6, g = (q >> 2) & 15, s = q & 3;
  int p = threadIdx.x;
  int ph = p >> 5, pw = p & 31;
  float cam = X[((size_t)(b * 4 + s) * NUM_GT + g) * NPAD + 1 + p];

  sb[p] = cam; __syncthreads();
  for (int off = 512; off > 0; off >>= 1) {
    if (p < off) sb[p] = fminf(sb[p], sb[p + off]);
    __syncthreads();
  }
  float vmin = sb[0]; __syncthreads();
  sb[p] = cam; __syncthreads();
  for (int off = 512; off > 0; off >>= 1) {
    if (p < off) sb[p] = fmaxf(sb[p], sb[p + off]);
    __syncthreads();
  }
  float vmax = sb[0]; __syncthreads();

  float camn = (cam - vmin) / (vmax - vmin + 1e-6f);
  out[(size_t)q * 1024 + p] = camn;
  sa[p] = (camn >= CAM_THR) ? 1.0f : 0.0f;
  __syncthreads();

  // dilate (zero pad is neutral for max of {0,1}) — separable
  float h = 0.0f;
  for (int dx = -5; dx <= 5; ++dx) { int x = pw + dx; if (x >= 0 && x < 32) h = fmaxf(h, sa[ph * 32 + x]); }
  sb[p] = h; __syncthreads();
  float d = 0.0f;
  for (int dy = -5; dy <= 5; ++dy) { int y = ph + dy; if (y >= 0 && y < 32) d = fmaxf(d, sb[y * 32 + pw]); }
  __syncthreads(); sa[p] = d; __syncthreads();

  // erode with OOB treated as 0 (reference zero-pads before min-pool) — separable
  float e = 3.0e38f;
  for (int dx = -5; dx <= 5; ++dx) { int x = pw + dx; e = fminf(e, (x >= 0 && x < 32) ? sa[ph * 32 + x] : 0.0f); }
  sb[p] = e; __syncthreads();
  float closed = 3.0e38f;
  for (int dy = -5; dy <= 5; ++dy) { int y = ph + dy; closed = fminf(closed, (y >= 0 && y < 32) ? sb[y * 32 + pw] : 0.0f); }
  out[131072 + (size_t)q * 1024 + p] = closed;
  __syncthreads(); sa[p] = closed; __syncthreads();

  if (p < 32) {                   // col_any[x]
    float ca = 0.0f;
    for (int y = 0; y < 32; ++y) ca = fmaxf(ca, sa[y * 32 + p]);
    sb[p] = ca;
  } else if (p < 64) {            // row_any[y]
    int y = p - 32; float ra = 0.0f;
    for (int x = 0; x < 32; ++x) ra = fmaxf(ra, sa[y * 32 + x]);
    sb[p] = ra;
  }
  __syncthreads();
  if (p == 0) {
    float xmin = 1.0e9f, xmax = -1.0e9f, ymin = 1.0e9f, ymax = -1.0e9f;
    bool any = false;
    for (int x = 0; x < 32; ++x)
      if (sb[x] > 0.5f) { any = true; xmin = fminf(xmin, (float)x); xmax = fmaxf(xmax, (float)x); }
    for (int y = 0; y < 32; ++y)
      if (sb[32 + y] > 0.5f) { ymin = fminf(ymin, (float)y); ymax = fmaxf(ymax, (float)y); }
    float* bb = out + 262144 + (size_t)q * 4;
    if (!any) { bb[0] = 0.0f; bb[1] = 0.0f; bb[2] = 1.0f; bb[3] = 1.0f; }
    else      { bb[0] = xmin; bb[1] = ymin; bb[2] = xmax; bb[3] = ymax; }
  }
}

extern "C" void kernel_launch(void* const* d_in, const int* in_sizes, int n_in,
                              void* d_out, int out_size, void* d_ws, size_t ws_size,
                              hipStream_t stream) {
  const float* attn = (const float*)d_in[0];   // (12,2,1125,1125) f32
  const int* pos = (const int*)d_in[1];        // (2,16) i32
  float* out = (float*)d_out;                  // camn|closed|bbox = 262656 f32
  float* ws = (float*)d_ws;
  float* invrs = ws;                                   // [2][4][1128]
  float* X = invrs + (size_t)Bsz * 4 * NPAD;           // [2][4][16][1128]
  float* H = X + (size_t)Bsz * 4 * NUM_GT * NPAD;      // [2][16][1128]

  k_rowsum<<<dim3(NPAD, 4, Bsz), 32, 0, stream>>>(attn, invrs);
  k_gather<<<dim3(NUM_GT, Bsz), 128, 0, stream>>>(attn, pos, invrs, X);

  k_scale<<<dim3(NUM_GT, Bsz), 128, 0, stream>>>(X, invrs, H, 3, 10);
  k_step<<<dim3(71, Bsz), 32, 0, stream>>>(attn, H, X, 10, 2);

  k_scale<<<dim3(NUM_GT, Bsz), 128, 0, stream>>>(X, invrs, H, 2, 9);
  k_step<<<dim3(71, Bsz), 32, 0, stream>>>(attn, H, X, 9, 1);

  k_scale<<<dim3(NUM_GT, Bsz), 128, 0, stream>>>(X, invrs, H, 1, 8);
  k_step<<<dim3(71, Bsz), 32, 0, stream>>>(attn, H, X, 8, 0);

  k_post<<<dim3(128), 1024, 0, stream>>>(X, out);
}